// CodeBook_40630390620792
// MI455X (gfx1250) — compile-verified
//
#include <hip/hip_runtime.h>
#include <hip/hip_bf16.h>

#define BATCH 16384
#define FDIM  65
#define MID   8
#define NCB   512
#define KPAD  68              // 65 padded to multiple of 4 for 16x16x4 WMMA
#define EPSF  1e-5f

typedef float v2f __attribute__((ext_vector_type(2)));
typedef float v8f __attribute__((ext_vector_type(8)));

// ---------------------------------------------------------------------------
// Kernel 0a: zero-padded codebook copy  cb[512][65] -> cbp[512][68]
// ---------------------------------------------------------------------------
__global__ void cbpad_kernel(const float* __restrict__ cb, float* __restrict__ cbp) {
    int i = blockIdx.x * blockDim.x + threadIdx.x;     // over 512*68
    if (i < NCB * KPAD) {
        int n = i / KPAD, k = i - n * KPAD;
        cbp[i] = (k < FDIM) ? cb[n * FDIM + k] : 0.0f;
    }
}

// ---------------------------------------------------------------------------
// Kernel 0b: codebook squared norms (512 values)
// ---------------------------------------------------------------------------
__global__ void cb2_kernel(const float* __restrict__ cb, float* __restrict__ cb2) {
    int n = blockIdx.x * blockDim.x + threadIdx.x;
    if (n < NCB) {
        float s = 0.f;
        #pragma unroll 5
        for (int k = 0; k < FDIM; ++k) {
            float v = cb[n * FDIM + k];
            s += v * v;
        }
        cb2[n] = s;
    }
}

// ---------------------------------------------------------------------------
// Kernel 1: encode — one thread per (b, f) position, produces 8 channels.
// Writes the "x" output region (also consumed as GEMM A-matrix by kernel 2).
// ---------------------------------------------------------------------------
__global__ void encode_kernel(const float* __restrict__ xk, const float* __restrict__ hl,
                              const float* __restrict__ c1w, const float* __restrict__ c1b,
                              const float* __restrict__ bng, const float* __restrict__ bnb,
                              const float* __restrict__ p1a,
                              const float* __restrict__ c2w, const float* __restrict__ c2b,
                              const float* __restrict__ lng, const float* __restrict__ lnb,
                              const float* __restrict__ p2a,
                              float* __restrict__ xout) {
    __shared__ float sW[209];
    const int tid = threadIdx.x;
    if (tid < 16)  sW[tid] = c1w[tid];                                   // conv1_w [8x2]
    if (tid < 8) {
        sW[16 + tid]  = c1b[tid];                                        // conv1_b
        sW[24 + tid]  = bng[tid] * rsqrtf(1.0f + EPSF);                  // BN scale
        sW[32 + tid]  = bnb[tid];                                        // BN beta
        sW[40 + tid]  = p1a[tid];                                        // PReLU1
        sW[192 + tid] = lng[tid];                                        // LN gamma
        sW[200 + tid] = lnb[tid];                                        // LN beta
    }
    if (tid < 128)                 sW[48 + tid]        = c2w[tid];       // conv2_w [16x8]
    else if (tid < 144)            sW[176 + tid - 128] = c2b[tid - 128]; // conv2_b
    if (tid == 144)                sW[208]             = p2a[0];         // PReLU2 scalar
    __syncthreads();

    const int e = blockIdx.x * blockDim.x + tid;
    if (e >= BATCH * FDIM) return;
    const int b = e / FDIM;
    const int f = e - b * FDIM;

    const float x0 = xk[e];
    const float x1 = hl[e];

    float h[MID];
    #pragma unroll
    for (int o = 0; o < MID; ++o) {
        float v = sW[o * 2] * x0 + sW[o * 2 + 1] * x1 + sW[16 + o];      // conv1
        v = v * sW[24 + o] + sW[32 + o];                                 // BN (inference)
        h[o] = (v >= 0.f) ? v : sW[40 + o] * v;                          // PReLU(MID)
    }

    float v8[MID];
    float mu = 0.f;
    #pragma unroll
    for (int o = 0; o < MID; ++o) {
        float a = sW[176 + o];
        float g = sW[176 + 8 + o];
        #pragma unroll
        for (int c = 0; c < MID; ++c) {
            a += sW[48 + o * MID + c] * h[c];                            // conv2 first half
            g += sW[48 + (o + 8) * MID + c] * h[c];                      // conv2 second half
        }
        float val = a * (1.0f / (1.0f + __expf(-g)));                    // GLU
        v8[o] = val;
        mu += val;
    }
    mu *= (1.0f / MID);
    float var = 0.f;
    #pragma unroll
    for (int o = 0; o < MID; ++o) {
        float d = v8[o] - mu;
        var += d * d;
    }
    var *= (1.0f / MID);
    const float rstd  = rsqrtf(var + EPSF);
    const float alpha = sW[208];
    #pragma unroll
    for (int o = 0; o < MID; ++o) {
        float t = (v8[o] - mu) * rstd * sW[192 + o] + sW[200 + o];       // LayerNorm(c)
        t = (t >= 0.f) ? t : alpha * t;                                  // PReLU scalar
        xout[(b * MID + o) * FDIM + f] = t;
    }
}

// ---------------------------------------------------------------------------
// Kernel 2: fused distance-GEMM (f32 WMMA 16x16x4) + argmin + gather.
// Block = 256 threads = 8 waves; block owns a 16-row M tile, wave owns 64 cols.
// All fragment loads are unconditional 8B loads (padded K=68), so there is no
// exec-mask divergence anywhere around the WMMAs (EXEC must be all-1s).
// ---------------------------------------------------------------------------
__global__ void __launch_bounds__(256)
vq_kernel(const float* __restrict__ x,     // [M=131072, 65]
          const float* __restrict__ cb,    // [512, 65]  (original, for gather)
          const float* __restrict__ cbp,   // [512, 68]  zero-padded
          const float* __restrict__ cb2,   // [512]
          float* __restrict__ x_out,       // [M, 65]
          float* __restrict__ x_cb) {      // [M, 65]
    const int tid  = threadIdx.x;
    const int lane = tid & 31;
    const int wave = tid >> 5;
    const int m0   = blockIdx.x * 16;

    __shared__ float sx[16 * KPAD];     // A tile, zero-padded K=65 -> 68
    __shared__ float smin[8 * 16];
    __shared__ int   sidx[8 * 16];
    __shared__ int   sfinal[16];

    for (int i = tid; i < 16 * KPAD; i += 256) {
        int r = i / KPAD, k = i - r * KPAD;
        sx[i] = (k < FDIM) ? x[(m0 + r) * FDIM + k] : 0.0f;
    }
    __syncthreads();

    const int nbase = wave * 64;
    const int l15   = lane & 15;
    const int hk    = (lane >> 4) ? 2 : 0;  // upper half-wave carries K+2,K+3

    // Per-lane base pointers (8B aligned: KPAD even, hk even)
    const float* arow = &sx[l15 * KPAD + hk];

    v8f acc0, acc1, acc2, acc3;
    #pragma unroll
    for (int j = 0; j < 8; ++j) { acc0[j] = 0.f; acc1[j] = 0.f; acc2[j] = 0.f; acc3[j] = 0.f; }

    const float* brow0 = &cbp[(nbase      + l15) * KPAD + hk];
    const float* brow1 = &cbp[(nbase + 16 + l15) * KPAD + hk];
    const float* brow2 = &cbp[(nbase + 32 + l15) * KPAD + hk];
    const float* brow3 = &cbp[(nbase + 48 + l15) * KPAD + hk];

    #pragma unroll
    for (int kk = 0; kk < KPAD / 4; ++kk) {
        const v2f a  = *(const v2f*)(arow  + kk * 4);   // ds_load_b64
        const v2f b0 = *(const v2f*)(brow0 + kk * 4);   // global_load_b64
        const v2f b1 = *(const v2f*)(brow1 + kk * 4);
        const v2f b2 = *(const v2f*)(brow2 + kk * 4);
        const v2f b3 = *(const v2f*)(brow3 + kk * 4);
        acc0 = __builtin_amdgcn_wmma_f32_16x16x4_f32(false, a, false, b0, (short)0, acc0, false, false);
        acc1 = __builtin_amdgcn_wmma_f32_16x16x4_f32(false, a, false, b1, (short)0, acc1, false, false);
        acc2 = __builtin_amdgcn_wmma_f32_16x16x4_f32(false, a, false, b2, (short)0, acc2, false, false);
        acc3 = __builtin_amdgcn_wmma_f32_16x16x4_f32(false, a, false, b3, (short)0, acc3, false, false);
    }

    // dist(m,n) = cb2[n] - 2*dot(x_m, cb_n)   (x^2 term constant per row -> dropped)
    float bestv[8];
    int   besti[8];
    #pragma unroll
    for (int j = 0; j < 8; ++j) { bestv[j] = 3.402823466e+38f; besti[j] = 0x7fffffff; }
    {
        const int n0 = nbase + l15;
        const float c20 = cb2[n0], c21 = cb2[n0 + 16], c22 = cb2[n0 + 32], c23 = cb2[n0 + 48];
        #pragma unroll
        for (int j = 0; j < 8; ++j) {
            float d0 = c20 - 2.0f * acc0[j];
            float d1 = c21 - 2.0f * acc1[j];
            float d2 = c22 - 2.0f * acc2[j];
            float d3 = c23 - 2.0f * acc3[j];
            float bv = d0; int bi = n0;
            if (d1 < bv) { bv = d1; bi = n0 + 16; }
            if (d2 < bv) { bv = d2; bi = n0 + 32; }
            if (d3 < bv) { bv = d3; bi = n0 + 48; }
            bestv[j] = bv; besti[j] = bi;
        }
    }
    // reduce across the 16 lanes of each half-wave (rows j / j+8)
    #pragma unroll
    for (int mask = 8; mask >= 1; mask >>= 1) {
        #pragma unroll
        for (int j = 0; j < 8; ++j) {
            float ov = __shfl_xor(bestv[j], mask, 32);
            int   oi = __shfl_xor(besti[j], mask, 32);
            if (ov < bestv[j] || (ov == bestv[j] && oi < besti[j])) { bestv[j] = ov; besti[j] = oi; }
        }
    }
    if (l15 == 0) {
        const int rbase = (lane >> 4) ? 8 : 0;
        #pragma unroll
        for (int j = 0; j < 8; ++j) {
            smin[wave * 16 + rbase + j] = bestv[j];
            sidx[wave * 16 + rbase + j] = besti[j];
        }
    }
    __syncthreads();
    if (tid < 16) {
        float bv = smin[tid];
        int   bi = sidx[tid];
        #pragma unroll
        for (int w = 1; w < 8; ++w) {
            float v = smin[w * 16 + tid];
            int   i = sidx[w * 16 + tid];
            if (v < bv || (v == bv && i < bi)) { bv = v; bi = i; }
        }
        sfinal[tid] = bi;
    }
    __syncthreads();

    // gather codebook rows; x_out == x_cb numerically (straight-through)
    for (int i = tid; i < 16 * FDIM; i += 256) {
        int r = i / FDIM, f = i - r * FDIM;
        float v = cb[sfinal[r] * FDIM + f];
        int o = (m0 + r) * FDIM + f;
        x_cb[o]  = v;
        x_out[o] = v;
    }
}

// ---------------------------------------------------------------------------
extern "C" void kernel_launch(void* const* d_in, const int* in_sizes, int n_in,
                              void* d_out, int out_size, void* d_ws, size_t ws_size,
                              hipStream_t stream) {
    const float* xk   = (const float*)d_in[0];
    const float* hl   = (const float*)d_in[1];
    const float* c1w  = (const float*)d_in[2];
    const float* c1b  = (const float*)d_in[3];
    const float* bng  = (const float*)d_in[4];
    const float* bnb  = (const float*)d_in[5];
    const float* p1a  = (const float*)d_in[6];
    const float* c2w  = (const float*)d_in[7];
    const float* c2b  = (const float*)d_in[8];
    const float* lng  = (const float*)d_in[9];
    const float* lnb  = (const float*)d_in[10];
    const float* p2a  = (const float*)d_in[11];
    const float* cbk  = (const float*)d_in[12];

    float* out = (float*)d_out;
    const int SZ = BATCH * MID * FDIM;          // 8,519,680
    float* x_out = out;                          // output 0: x_out (== x_cb)
    float* x     = out + SZ;                     // output 1: x (encoder result)
    float* x_cb  = out + 2 * SZ;                 // output 2: x_cb

    float* cbp = (float*)d_ws;                   // [512*68] padded codebook
    float* cb2 = cbp + NCB * KPAD;               // [512]

    cbpad_kernel<<<(NCB * KPAD + 255) / 256, 256, 0, stream>>>(cbk, cbp);
    cb2_kernel<<<2, 256, 0, stream>>>(cbk, cb2);

    const int npos = BATCH * FDIM;               // 1,064,960 -> 4160 blocks
    encode_kernel<<<(npos + 255) / 256, 256, 0, stream>>>(
        xk, hl, c1w, c1b, bng, bnb, p1a, c2w, c2b, lng, lnb, p2a, x);

    const int M = BATCH * MID;                   // 131072 rows -> 8192 tiles of 16
    vq_kernel<<<M / 16, 256, 0, stream>>>(x, cbk, cbp, cb2, x_out, x_cb);
}